// EPDetector_55018531062641
// MI455X (gfx1250) — compile-verified
//
#include <hip/hip_runtime.h>
#include <math.h>

typedef float v2f __attribute__((ext_vector_type(2)));
typedef float v8f __attribute__((ext_vector_type(8)));

#define WAVES_PER_BLOCK 4
#define LDS_PER_WAVE    2193
// per-wave LDS float offsets
#define SR_OFF   0      // 32x33 real rep of S -> Cholesky L32 in place
#define Q_OFF    1056   // 17x33 column-major RHS -> Q = L32^{-1} RHS in place
#define HTH_OFF  1617   // 16x17 Gram matrix Q_h^T Q_h
#define A_OFF    1889   // 16x17 EP working matrix / Cholesky L16
#define OUT_OFF  2161   // 32 staged outputs

__launch_bounds__(128, 2)
__global__ void ep_detect_kernel(const float* __restrict__ y_re, const float* __restrict__ y_im,
                                 const float* __restrict__ h_re, const float* __restrict__ h_im,
                                 const float* __restrict__ s_re, const float* __restrict__ s_im,
                                 float* __restrict__ out, int B)
{
    __shared__ float lds[WAVES_PER_BLOCK * LDS_PER_WAVE];
    const int lane = threadIdx.x & 31;
    const int wave = threadIdx.x >> 5;
    const int b = blockIdx.x * WAVES_PER_BLOCK + wave;
    if (b >= B) return;                       // wave-uniform exit

    float* w   = lds + wave * LDS_PER_WAVE;
    float* sr  = w + SR_OFF;
    float* q   = w + Q_OFF;
    float* hth = w + HTH_OFF;
    float* am  = w + A_OFF;
    float* ob  = w + OUT_OFF;

    // ---------- load Sr = [[Re S, -Im S],[Im S, Re S]] and RHS = [rep(h) | vec(y)] ----------
    {
        const float* sre = s_re + (size_t)b * 256;
        const float* sim = s_im + (size_t)b * 256;
        for (int idx = lane; idx < 256; idx += 32) {
            int i = idx >> 4, j = idx & 15;
            float re = sre[idx], im = sim[idx];
            sr[i*33 + j]            = re;
            sr[(16+i)*33 + 16 + j]  = re;
            sr[(16+i)*33 + j]       = im;
            sr[i*33 + 16 + j]       = -im;
        }
        const float* hre = h_re + (size_t)b * 128;
        const float* him = h_im + (size_t)b * 128;
        for (int idx = lane; idx < 128; idx += 32) {
            int m = idx >> 3, s = idx & 7;
            float re = hre[idx], im = him[idx];
            q[s*33 + m]           = re;     // col s     : [Re ; Im]
            q[s*33 + 16 + m]      = im;
            q[(8+s)*33 + m]       = -im;    // col 8+s   : [-Im ; Re]
            q[(8+s)*33 + 16 + m]  = re;
        }
        if (lane < 16) {
            q[16*33 + lane]      = y_re[(size_t)b*16 + lane];
            q[16*33 + 16 + lane] = y_im[(size_t)b*16 + lane];
        }
    }
    __builtin_amdgcn_wave_barrier();

    // ---------- Cholesky of Sr (32x32 SPD), right-looking, lane = row ----------
    for (int j = 0; j < 32; ++j) {
        float d  = sqrtf(sr[j*33 + j]);      // broadcast read
        float rd = 1.0f / d;
        if (lane == j) sr[j*33 + j] = d;
        float lij = 0.0f;
        if (lane > j) { lij = sr[lane*33 + j] * rd; sr[lane*33 + j] = lij; }
        __builtin_amdgcn_wave_barrier();
        for (int k = j + 1; k < 32; ++k) {
            float lkj = sr[k*33 + j];        // broadcast read
            if (lane >= k) sr[lane*33 + k] -= lij * lkj;
        }
        __builtin_amdgcn_wave_barrier();
    }

    // ---------- forward substitution: Q = L32^{-1} RHS, lane = column (17 cols) ----------
    if (lane < 17) {
        float* col = q + lane*33;
        for (int i = 0; i < 32; ++i) {
            float s = col[i];
            const float* Lr = sr + i*33;
            for (int k = 0; k < i; ++k) s -= Lr[k] * col[k];
            col[i] = s / Lr[i];
        }
    }
    __builtin_amdgcn_wave_barrier();

    // ---------- h_t_y (lane s = component, lanes 16..31 mirror) ----------
    const int s = lane & 15;
    float hty = 0.0f;
    {
        const float* yc = q + 16*33;
        const float* qc = q + s*33;
        for (int r = 0; r < 32; ++r) hty += qc[r] * yc[r];
    }

    // ---------- h_t_h = Q_h^T Q_h via 8x V_WMMA_F32_16X16X4_F32 ----------
    {
        const int half = lane >> 4;
        const int lr   = lane & 15;
        v8f acc = {0.f,0.f,0.f,0.f,0.f,0.f,0.f,0.f};
        #pragma unroll
        for (int c = 0; c < 8; ++c) {
            // A = (Q_h^T)[:,4c..4c+3] and B = Q_h[4c..4c+3,:] share one register image
            int base = lr*33 + 4*c + 2*half;
            v2f ab;
            ab.x = q[base];
            ab.y = q[base + 1];
            acc = __builtin_amdgcn_wmma_f32_16x16x4_f32(
                      false, ab, false, ab, (short)0, acc, false, false);
        }
        __builtin_amdgcn_wave_barrier();
        #pragma unroll
        for (int r = 0; r < 8; ++r)
            hth[(r + 8*half)*17 + lr] = acc[r];   // D: VGPR r = row r (+8 for upper half)
    }
    __builtin_amdgcn_wave_barrier();

    // ---------- EP loop: lane s owns real component s ----------
    const float no = 0.5f;
    const float P0c =  0.22360680f, P1c =  0.67082039f;
    const float P2c = -0.22360680f, P3c = -0.67082039f;
    float lam = 4.0f, gam = 0.0f;                 // 1/ES, ES = 0.25
    float lg0 = 0.f, lg1 = 0.f, lg2 = 0.f, lg3 = 0.f;

    for (int it = 0; it < 10; ++it) {
        // A = h_t_h + no*lam*I
        for (int t = 0; t < 8; ++t) {
            int idx = lane + 32*t;
            int ij  = (idx >> 4)*17 + (idx & 15);
            am[ij] = hth[ij];
        }
        __builtin_amdgcn_wave_barrier();
        if (lane < 16) am[s*17 + s] += no * lam;
        __builtin_amdgcn_wave_barrier();

        // Cholesky 16x16
        for (int j = 0; j < 16; ++j) {
            float d  = sqrtf(am[j*17 + j]);
            float rd = 1.0f / d;
            if (lane == j) am[j*17 + j] = d;
            float lij = 0.0f;
            if (lane > j && lane < 16) { lij = am[lane*17 + j] * rd; am[lane*17 + j] = lij; }
            __builtin_amdgcn_wave_barrier();
            for (int k = j + 1; k < 16; ++k) {
                float lkj = am[k*17 + j];
                if (lane >= k && lane < 16) am[lane*17 + k] -= lij * lkj;
            }
            __builtin_amdgcn_wave_barrier();
        }

        // column s of M = L16^{-1} (uniform-address broadcast reads, unrolled)
        float x[16];
        #pragma unroll
        for (int k = 0; k < 16; ++k) {
            float a2 = (k == s) ? 1.0f : 0.0f;
            #pragma unroll
            for (int m = 0; m < k; ++m) a2 -= am[k*17 + m] * x[m];
            x[k] = a2 / am[k*17 + k];
        }

        // sigma_s = no * diag(A^{-1})_s = no * |col_s(M)|^2
        float dsum = 0.f;
        #pragma unroll
        for (int k = 0; k < 16; ++k) dsum += x[k]*x[k];
        float sigma = no * dsum;

        // mu = A^{-1} rhs = M^T (M rhs)  — cross-lane sum within 16-lane groups
        float rhs = hty + no * gam;
        float t16[16];
        #pragma unroll
        for (int k = 0; k < 16; ++k) t16[k] = x[k] * rhs;
        #pragma unroll
        for (int off = 1; off <= 8; off <<= 1) {
            #pragma unroll
            for (int k = 0; k < 16; ++k) t16[k] += __shfl_xor(t16[k], off, 32);
        }
        float mu = 0.f;
        #pragma unroll
        for (int k = 0; k < 16; ++k) mu += x[k] * t16[k];

        // EP moment matching (per-lane scalar)
        float v_obs = fmaxf(1.0f / (1.0f/sigma - lam), 1e-6f);
        float x_obs = v_obs * (mu/sigma - gam);
        float inv2v = 0.5f / v_obs;
        lg0 = -(x_obs - P0c)*(x_obs - P0c)*inv2v;
        lg1 = -(x_obs - P1c)*(x_obs - P1c)*inv2v;
        lg2 = -(x_obs - P2c)*(x_obs - P2c)*inv2v;
        lg3 = -(x_obs - P3c)*(x_obs - P3c)*inv2v;
        float mx = fmaxf(fmaxf(lg0, lg1), fmaxf(lg2, lg3));
        float e0 = expf(lg0 - mx), e1 = expf(lg1 - mx);
        float e2 = expf(lg2 - mx), e3 = expf(lg3 - mx);
        float rz = 1.0f / (e0 + e1 + e2 + e3);
        float xm = (P0c*e0 + P1c*e1 + P2c*e2 + P3c*e3) * rz;
        float vv = ((P0c-xm)*(P0c-xm)*e0 + (P1c-xm)*(P1c-xm)*e1 +
                    (P2c-xm)*(P2c-xm)*e2 + (P3c-xm)*(P3c-xm)*e3) * rz;
        vv = fmaxf(vv, 1e-6f);
        float lam_c = 1.0f/vv - 1.0f/v_obs;
        float gam_c = xm/vv - x_obs/v_obs;
        bool  neg   = lam_c < 0.0f;
        float lam_n = neg ? lam : lam_c;
        float gam_n = neg ? gam : gam_c;
        lam = 0.1f*lam_n + 0.9f*lam;          // BETA = 0.9 damping
        gam = 0.1f*gam_n + 0.9f*gam;
    }

    // ---------- max-log LLRs (K=2 Gray PAM: bit0 -> {2,3}|{0,1}, bit1 -> {1,3}|{0,2}) ----------
    float bb0 = fmaxf(lg2, lg3) - fmaxf(lg0, lg1);
    float bb1 = fmaxf(lg1, lg3) - fmaxf(lg0, lg2);
    __builtin_amdgcn_wave_barrier();
    if (lane < 8) {                // real components -> llr1 at slots 0,2
        ob[4*lane + 0] = bb0;
        ob[4*lane + 2] = bb1;
    } else if (lane < 16) {        // imag components -> llr2 at slots 1,3
        ob[4*(lane-8) + 1] = bb0;
        ob[4*(lane-8) + 3] = bb1;
    }
    __builtin_amdgcn_wave_barrier();
    out[(size_t)b*32 + lane] = ob[lane];
}

extern "C" void kernel_launch(void* const* d_in, const int* in_sizes, int n_in,
                              void* d_out, int out_size, void* d_ws, size_t ws_size,
                              hipStream_t stream) {
    const float* y_re = (const float*)d_in[0];
    const float* y_im = (const float*)d_in[1];
    const float* h_re = (const float*)d_in[2];
    const float* h_im = (const float*)d_in[3];
    const float* s_re = (const float*)d_in[4];
    const float* s_im = (const float*)d_in[5];
    float* out = (float*)d_out;
    int B = in_sizes[0] / 16;                              // y_re is (B,16)
    dim3 block(32 * WAVES_PER_BLOCK);
    dim3 grid((B + WAVES_PER_BLOCK - 1) / WAVES_PER_BLOCK);
    ep_detect_kernel<<<grid, block, 0, stream>>>(y_re, y_im, h_re, h_im, s_re, s_im, out, B);
}